// GOTSim_50766513439322
// MI455X (gfx1250) — compile-verified
//
#include <hip/hip_runtime.h>
#include <math.h>

// Problem constants (from the reference)
#define N1v 512
#define N2v 448
#define DFv 512
#define HFv 512
#define E1v 8192
#define E2v 7168
#define NTOT 960
#define BIGC 99999.0f

typedef __attribute__((ext_vector_type(2))) float v2f;
typedef __attribute__((ext_vector_type(8))) float v8f;

// ---------------- degree / symmetric normalization ----------------
__global__ void deg_init_k(float* deg, int n) {
  int i = blockIdx.x * blockDim.x + threadIdx.x;
  if (i < n) deg[i] = 1.0f;  // self loop
}
__global__ void deg_acc_k(const int* __restrict__ dst, float* deg, int e) {
  int i = blockIdx.x * blockDim.x + threadIdx.x;
  if (i < e) atomicAdd(&deg[dst[i]], 1.0f);
}
__global__ void deg_fin_k(float* deg, int n) {
  int i = blockIdx.x * blockDim.x + threadIdx.x;
  if (i < n) deg[i] = rsqrtf(fmaxf(deg[i], 1.0f));
}

// ---------------- fp32 WMMA GEMM: C[MxN] = op(A) @ B ----------------
// A: MxK row-major, B: KxN row-major, C: MxN row-major. M,N multiples of 16, K mult of 4.
// One wave computes one 16x16 f32 tile via V_WMMA_F32_16X16X4_F32 (K stepped by 4).
// RELU_A templated so layer-0 has no max and layers 1/2 have no runtime select.
template <int RELU_A>
__global__ void __launch_bounds__(256) wmma_gemm_nn(
    const float* __restrict__ A, const float* __restrict__ B, float* __restrict__ C,
    int M, int N, int K) {
  int wave = (blockIdx.x * blockDim.x + threadIdx.x) >> 5;
  int lane = threadIdx.x & 31;
  int tilesN = N >> 4, tilesM = M >> 4;
  if (wave >= tilesM * tilesN) return;           // uniform per wave: EXEC stays all-1s
  int tm = wave / tilesN, tn = wave - tm * tilesN;
  int half = lane >> 4, col = lane & 15;
  int arow = tm * 16 + col;                      // A: M = lane%16 for both halves
  int bcol = tn * 16 + col;                      // B/C/D: N = lane%16
  const v2f* __restrict__ Arow2 = (const v2f*)(A + (size_t)arow * K);  // 8B-aligned pairs
  v8f acc = {0.f, 0.f, 0.f, 0.f, 0.f, 0.f, 0.f, 0.f};
  for (int k0 = 0; k0 < K; k0 += 4) {
    int kk = k0 + half * 2;                      // lanes 0-15: K=k0,k0+1 ; 16-31: K=k0+2,k0+3
    v2f a = Arow2[kk >> 1];                      // global_load_b64
    if (RELU_A) { a[0] = fmaxf(a[0], 0.f); a[1] = fmaxf(a[1], 0.f); }
    v2f b;
    b[0] = B[(size_t)kk * N + bcol];             // lane-coalesced (16 contiguous floats)
    b[1] = B[(size_t)(kk + 1) * N + bcol];
    acc = __builtin_amdgcn_wmma_f32_16x16x4_f32(false, a, false, b, (short)0, acc,
                                                false, false);
  }
#pragma unroll
  for (int r = 0; r < 8; r++)                    // C VGPR r: M=r (lanes 0-15) / M=8+r
    C[(size_t)(tm * 16 + half * 8 + r) * N + tn * 16 + col] = acc[r];
}

// C[MxN](ldc) = -(A @ Bt^T): A MxK row-major, Bt NxK row-major (x2), negate, strided store
__global__ void __launch_bounds__(256) wmma_gemm_nt_neg(
    const float* __restrict__ A, const float* __restrict__ Bt, float* __restrict__ C,
    int M, int N, int K, int ldc) {
  int wave = (blockIdx.x * blockDim.x + threadIdx.x) >> 5;
  int lane = threadIdx.x & 31;
  int tilesN = N >> 4, tilesM = M >> 4;
  if (wave >= tilesM * tilesN) return;
  int tm = wave / tilesN, tn = wave - tm * tilesN;
  int half = lane >> 4, col = lane & 15;
  int arow = tm * 16 + col;
  int brow = tn * 16 + col;
  const v2f* __restrict__ Arow2 = (const v2f*)(A + (size_t)arow * K);
  const v2f* __restrict__ Brow2 = (const v2f*)(Bt + (size_t)brow * K);
  v8f acc = {0.f, 0.f, 0.f, 0.f, 0.f, 0.f, 0.f, 0.f};
  for (int k0 = 0; k0 < K; k0 += 4) {
    int kk = k0 + half * 2;
    v2f a = Arow2[kk >> 1];                      // global_load_b64
    v2f b = Brow2[kk >> 1];                      // B_eff[k][n] = x2[n][k], global_load_b64
    acc = __builtin_amdgcn_wmma_f32_16x16x4_f32(false, a, false, b, (short)0, acc,
                                                false, false);
  }
#pragma unroll
  for (int r = 0; r < 8; r++)
    C[(size_t)(tm * 16 + half * 8 + r) * ldc + tn * 16 + col] = -acc[r];
}

// ---------------- GCN aggregation ----------------
// out = dinv^2 * xw (self loop) + bias, then edges atomically accumulated
__global__ void agg_init_k(const float* __restrict__ xw, const float* __restrict__ dinv,
                           const float* __restrict__ bias, float* __restrict__ out,
                           int n, int H) {
  int idx = blockIdx.x * blockDim.x + threadIdx.x;
  if (idx >= n * H) return;
  int row = idx / H, h = idx - row * H;
  float di = dinv[row];
  out[idx] = di * di * xw[idx] + bias[h];
}
__global__ void agg_edges_k(const float* __restrict__ xw, const float* __restrict__ dinv,
                            const int* __restrict__ src, const int* __restrict__ dst,
                            float* __restrict__ out, int H) {
  int e = blockIdx.x;
  int s = src[e], d = dst[e];
  float norm = dinv[s] * dinv[d];
  for (int h = threadIdx.x; h < H; h += blockDim.x)
    atomicAdd(&out[d * H + h], norm * xw[s * H + h]);
}

// ---------------- diag dot products and cost-matrix borders ----------------
__global__ void rowdot_k(const float* __restrict__ X, const float* __restrict__ w,
                         float* __restrict__ out, int K) {
  __shared__ float red[256];
  int row = blockIdx.x;
  float s = 0.f;
  for (int k = threadIdx.x; k < K; k += 256) s += X[row * K + k] * w[k];
  red[threadIdx.x] = s;
  __syncthreads();
  for (int off = 128; off > 0; off >>= 1) {
    if (threadIdx.x < off) red[threadIdx.x] += red[threadIdx.x + off];
    __syncthreads();
  }
  if (threadIdx.x == 0) out[row] = red[0];
}

__global__ void fill_borders_k(float* __restrict__ Cm, const float* __restrict__ d1,
                               const float* __restrict__ d2) {
  int idx = blockIdx.x * blockDim.x + threadIdx.x;
  if (idx >= NTOT * NTOT) return;
  int i = idx / NTOT, j = idx - i * NTOT;
  if (i < N1v && j < N2v) return;  // main block filled by wmma_gemm_nt_neg
  float val;
  if (i < N1v) {                   // Del block [N1 x N1] at (0, N2)
    int jj = j - N2v;
    val = (i == jj) ? -d1[i] : BIGC;
  } else if (j < N2v) {            // Ins block [N2 x N2] at (N1, 0)
    int ii = i - N1v;
    val = (ii == j) ? -d2[ii] : BIGC;
  } else {                         // dummy zeros
    val = 0.f;
  }
  Cm[idx] = val;
}

// ---------------- Jonker-Volgenant LAP, one LDS-resident workgroup per layer ----------------
__global__ void __launch_bounds__(1024) lap_k(const float* __restrict__ Cms,
                                              float* __restrict__ losses) {
  const float* Cm = Cms + (size_t)blockIdx.x * NTOT * NTOT;
  __shared__ double u[NTOT + 1], v[NTOT + 1], minv[NTOT + 1];
  __shared__ int p[NTOT + 1], way[NTOT + 1];
  __shared__ unsigned char used[NTOT + 1];
  __shared__ double redv[1024];
  __shared__ int redi[1024];
  __shared__ int s_j0;
  const double INFD = 1e18;
  int tid = threadIdx.x;

  for (int j = tid; j <= NTOT; j += 1024) { u[j] = 0.0; v[j] = 0.0; p[j] = 0; way[j] = 0; }
  __syncthreads();

  for (int i = 1; i <= NTOT; i++) {
    if (tid == 0) { p[0] = i; s_j0 = 0; }
    for (int j = tid; j <= NTOT; j += 1024) { minv[j] = INFD; used[j] = 0; }
    __syncthreads();
    while (true) {
      int j0 = s_j0;
      if (tid == 0) used[j0] = 1;
      __syncthreads();
      int i0 = p[j0];
      double u0 = u[i0];
      // relax all free columns + local argmin (each column owned by one thread)
      double bestv = INFD;
      int bestj = 0;
      for (int j = tid + 1; j <= NTOT; j += 1024) {
        if (!used[j]) {
          double cur = (double)Cm[(i0 - 1) * NTOT + (j - 1)] - u0 - v[j];
          if (cur < minv[j]) { minv[j] = cur; way[j] = j0; }
          if (minv[j] < bestv) { bestv = minv[j]; bestj = j; }
        }
      }
      redv[tid] = bestv;
      redi[tid] = bestj;
      __syncthreads();
      for (int off = 512; off > 0; off >>= 1) {
        if (tid < off && redv[tid + off] < redv[tid]) {
          redv[tid] = redv[tid + off];
          redi[tid] = redi[tid + off];
        }
        __syncthreads();
      }
      int j1 = redi[0];
      double delta = redv[0];
      // dual update (rows p[j] for used j are distinct -> no write conflicts)
      for (int j = tid; j <= NTOT; j += 1024) {
        if (used[j]) { u[p[j]] += delta; v[j] -= delta; }
        else minv[j] -= delta;
      }
      if (tid == 0) s_j0 = j1;
      __syncthreads();
      if (p[s_j0] == 0) break;
    }
    if (tid == 0) {  // augment
      int j0v = s_j0;
      while (j0v) { int j1v = way[j0v]; p[j0v] = p[j1v]; j0v = j1v; }
    }
    __syncthreads();
  }
  // loss = mean_r C[r, col[r]] = (1/n) * sum_j C[p[j]-1, j-1]
  double s = 0.0;
  for (int j = tid + 1; j <= NTOT; j += 1024) s += (double)Cm[(p[j] - 1) * NTOT + (j - 1)];
  redv[tid] = s;
  __syncthreads();
  for (int off = 512; off > 0; off >>= 1) {
    if (tid < off) redv[tid] += redv[tid + off];
    __syncthreads();
  }
  if (tid == 0) losses[blockIdx.x] = (float)(redv[0] / (double)NTOT);
}

// ---------------- final linear + sigmoid ----------------
__global__ void finalize_k(const float* __restrict__ losses, const float* __restrict__ ot_w,
                           const float* __restrict__ ot_b, float* __restrict__ out) {
  if (threadIdx.x == 0 && blockIdx.x == 0) {
    float logit = ot_b[0] + losses[0] * ot_w[0] + losses[1] * ot_w[1] + losses[2] * ot_w[2];
    out[1] = logit;
    out[0] = 1.f / (1.f + expf(-logit));
  }
}

extern "C" void kernel_launch(void* const* d_in, const int* in_sizes, int n_in,
                              void* d_out, int out_size, void* d_ws, size_t ws_size,
                              hipStream_t stream) {
  (void)in_sizes; (void)n_in; (void)out_size; (void)ws_size;
  const int* ei1 = (const int*)d_in[0];
  const int* ei2 = (const int*)d_in[1];
  const float* x1 = (const float*)d_in[2];
  const float* x2 = (const float*)d_in[3];
  const float* Wm[3] = {(const float*)d_in[4], (const float*)d_in[6], (const float*)d_in[8]};
  const float* Bm[3] = {(const float*)d_in[5], (const float*)d_in[7], (const float*)d_in[9]};
  const float* insp = (const float*)d_in[10];  // [3, H]
  const float* delp = (const float*)d_in[11];  // [3, H]
  const float* otw  = (const float*)d_in[12];  // [3]
  const float* otb  = (const float*)d_in[13];  // [1]
  float* out = (float*)d_out;

  // workspace layout (floats)
  float* ws = (float*)d_ws;
  float* dinv1 = ws;                            // 512
  float* dinv2 = ws + 512;                      // 448
  float* xw    = ws + 960;                      // 512*512 (shared scratch)
  float* f1    = xw + N1v * HFv;                // 3 * 512*512 (pre-ReLU feats g1)
  float* f2    = f1 + 3 * N1v * HFv;            // 3 * 448*512 (pre-ReLU feats g2)
  float* diag1 = f2 + 3 * N2v * HFv;            // 512
  float* diag2 = diag1 + N1v;                   // 448
  float* losses = diag2 + N2v;                  // 3 (+pad)
  float* Cms   = losses + 8;                    // 3 * 960*960

  const int* srcs[2] = {ei1, ei2 };
  const int* dsts[2] = {ei1 + E1v, ei2 + E2v};
  const float* xin[2] = {x1, x2};
  float* dinv[2] = {dinv1, dinv2};
  float* feats[2] = {f1, f2};
  const int nn[2] = {N1v, N2v};
  const int ee[2] = {E1v, E2v};

  for (int g = 0; g < 2; g++) {
    int n = nn[g], e = ee[g];
    deg_init_k<<<(n + 255) / 256, 256, 0, stream>>>(dinv[g], n);
    deg_acc_k<<<(e + 255) / 256, 256, 0, stream>>>(dsts[g], dinv[g], e);
    deg_fin_k<<<(n + 255) / 256, 256, 0, stream>>>(dinv[g], n);

    const float* in = xin[g];
    for (int l = 0; l < 3; l++) {
      float* fl = feats[g] + (size_t)l * n * HFv;
      int tiles = (n >> 4) * (HFv >> 4);
      int blks = (tiles + 7) / 8;
      if (l == 0)
        wmma_gemm_nn<0><<<blks, 256, 0, stream>>>(in, Wm[l], xw, n, HFv, DFv);
      else
        wmma_gemm_nn<1><<<blks, 256, 0, stream>>>(in, Wm[l], xw, n, HFv, DFv);
      agg_init_k<<<(n * HFv + 255) / 256, 256, 0, stream>>>(xw, dinv[g], Bm[l], fl, n, HFv);
      agg_edges_k<<<e, 256, 0, stream>>>(xw, dinv[g], srcs[g], dsts[g], fl, HFv);
      in = fl;  // next layer reads pre-ReLU feats; GEMM applies ReLU on load
    }
  }

  // build the three 960x960 cost matrices
  for (int l = 0; l < 3; l++) {
    const float* f1l = f1 + (size_t)l * N1v * HFv;
    const float* f2l = f2 + (size_t)l * N2v * HFv;
    float* Cl = Cms + (size_t)l * NTOT * NTOT;
    rowdot_k<<<N1v, 256, 0, stream>>>(f1l, delp + l * HFv, diag1, HFv);
    rowdot_k<<<N2v, 256, 0, stream>>>(f2l, insp + l * HFv, diag2, HFv);
    fill_borders_k<<<(NTOT * NTOT + 255) / 256, 256, 0, stream>>>(Cl, diag1, diag2);
    int tiles = (N1v >> 4) * (N2v >> 4);
    wmma_gemm_nt_neg<<<(tiles + 7) / 8, 256, 0, stream>>>(f1l, f2l, Cl, N1v, N2v, HFv, NTOT);
  }

  // three concurrent LAP solves (one WGP-resident workgroup each), then the head
  lap_k<<<3, 1024, 0, stream>>>(Cms, losses);
  finalize_k<<<1, 32, 0, stream>>>(losses, otw, otb, out);
}